// EarthMoverDistanceConfidence_15899968929944
// MI455X (gfx1250) — compile-verified
//
#include <hip/hip_runtime.h>
#include <math.h>

// EMD approxmatch + matchcost for b=8, n=m=2048, 3-D points (Fan et al.)
// Distances via V_WMMA_F32_16X16X4_F32; exp2 on the TRANS pipe; match in HBM.

#define B_      8
#define N_      2048
#define BN_     (B_ * N_)
#define TILES_  (N_ / 16)        // 128 16-wide tiles per side
#define PASSBLK (B_ * TILES_)    // 1024 blocks; 8 waves/block split the j-range

typedef __attribute__((ext_vector_type(2))) float v2f;
typedef __attribute__((ext_vector_type(8))) float v8f;

#define LOG2E_ 1.44269504088896340736f
#define HSKEW  (N_ + 16)         // half-select stride w/ bank skew

__device__ __forceinline__ float fast_exp2(float x) {
#if __has_builtin(__builtin_amdgcn_exp2f)
  return __builtin_amdgcn_exp2f(x);   // raw v_exp_f32
#else
  return exp2f(x);
#endif
}

// Sum within each 16-lane half of a wave32 (xor masks < 16 never cross halves).
__device__ __forceinline__ float half16_sum(float v) {
  v += __shfl_xor(v, 8, 32);
  v += __shfl_xor(v, 4, 32);
  v += __shfl_xor(v, 2, 32);
  v += __shfl_xor(v, 1, 32);
  return v;
}

__global__ void emd_init(float* __restrict__ cost,
                         float* __restrict__ remL,
                         float* __restrict__ remR) {
  int idx = blockIdx.x * blockDim.x + threadIdx.x;
  if (idx < BN_) { remL[idx] = 1.0f; remR[idx] = 1.0f; }  // multiL = multiR = 1
  if (idx < B_)  cost[idx] = 0.0f;
}

// Generic "ratio" pass: for every row x of side X,
//   s_x = sum_y exp2(l2 * ||X_x - Y_y||^2) * vec_y       (l2 = level * log2(e))
//   ratioX_x = remX_x / (1e-9 + s_x)
// optionally remXout_x = max(0, remX_x - ratioX_x * s_x)  (pass-2 remR update)
__global__ __launch_bounds__(256)
void emd_ratio(const float* __restrict__ X, const float* __restrict__ Y,
               const float* __restrict__ vec, const float* __restrict__ remX,
               float* __restrict__ ratioX, float* __restrict__ remXout,
               float l2) {
  // Pre-swizzled WMMA B-operands: index = j + h*HSKEW (h = lane>>4)
  __shared__ float sB0[2 * HSKEW];   // h=0: qx  | h=1: qz
  __shared__ float sB1[2 * HSKEW];   // h=0: qy  | h=1: 1.0
  __shared__ float sYs[N_];          // |q|^2
  __shared__ float sYv[N_];          // vec
  __shared__ float sRed[8][16];

  const int b  = blockIdx.x / TILES_;
  const int i0 = (blockIdx.x % TILES_) * 16;
  {
    const size_t yb = (size_t)b * N_ * 3;
    for (int t = threadIdx.x; t < N_; t += 256) {
      float x = Y[yb + 3 * t], y = Y[yb + 3 * t + 1], z = Y[yb + 3 * t + 2];
      sB0[t] = x; sB0[HSKEW + t] = z;
      sB1[t] = y; sB1[HSKEW + t] = 1.0f;
      sYs[t] = x * x + y * y + z * z;
      sYv[t] = vec[b * N_ + t];
    }
  }
  __syncthreads();

  const int lane = threadIdx.x & 31;
  const int w    = __builtin_amdgcn_readfirstlane(threadIdx.x >> 5);
  const int r    = lane & 15, h = lane >> 4;
  const int hoff = h * HSKEW;

  // A operand (16x4 f32): lane = M, (vgpr,half) -> K = vgpr + 2*half
  // A row = (-2px, -2py, -2pz, |p|^2)
  const size_t xb = ((size_t)b * N_ + (i0 + r)) * 3;
  const float px = X[xb], py = X[xb + 1], pz = X[xb + 2];
  const float ssx = px * px + py * py + pz * pz;
  v2f av; av.x = h ? -2.f * pz : -2.f * px;
          av.y = h ?  ssx      : -2.f * py;

  v8f acc = {};
  const v8f cz = {};
  for (int s = 0; s < 16; ++s) {               // uniform trip count
    const int jn = w * 16 + s * 128 + r;
    v2f bv; bv.x = sB0[hoff + jn];             // unconditional ds loads
            bv.y = sB1[hoff + jn];
    v8f d = __builtin_amdgcn_wmma_f32_16x16x4_f32(
        false, av, false, bv, (short)0, cz, false, false);
    const float l2sq2 = l2 * sYs[jn];          // hoisted per lane per tile
    const float vv    = sYv[jn];
#pragma unroll
    for (int k = 0; k < 8; ++k) {
      // exp(level*sq) = exp2(l2*d[k] + l2*|q|^2)  -> fma + v_exp + pk_fma
      acc[k] += fast_exp2(__builtin_fmaf(l2, d[k], l2sq2)) * vv;
    }
  }
#pragma unroll
  for (int k = 0; k < 8; ++k) acc[k] = half16_sum(acc[k]);
  if (r == 0) {
#pragma unroll
    for (int k = 0; k < 8; ++k) sRed[w][k + 8 * h] = acc[k];
  }
  __syncthreads();

  if (threadIdx.x < 16) {
    const int row = threadIdx.x;
    float rs = 0.f;
#pragma unroll
    for (int q = 0; q < 8; ++q) rs += sRed[q][row];
    const int i = b * N_ + i0 + row;
    const float rx  = remX[i];
    const float rat = rx / (1e-9f + rs);
    ratioX[i] = rat;
    if (remXout) remXout[i] = fmaxf(0.f, rx - rat * rs);
  }
}

// Pass 3: delta = W * ratioL_i * ratioR_j; match += delta (layout b,i,j);
// remL_i = max(0, remL_i - sum_j delta); last level also accumulates cost.
__global__ __launch_bounds__(256)
void emd_update(const float* __restrict__ X, const float* __restrict__ Y,
                const float* __restrict__ ratioL, const float* __restrict__ ratioR,
                float* __restrict__ remL, float* __restrict__ match,
                float* __restrict__ cost, float l2, int first, int last) {
  __shared__ float sB0[2 * HSKEW];
  __shared__ float sB1[2 * HSKEW];
  __shared__ float sYs[N_];
  __shared__ float sYv[N_];
  __shared__ float sRed[8][16];
  __shared__ float sCost[8];

  const int b  = blockIdx.x / TILES_;
  const int i0 = (blockIdx.x % TILES_) * 16;
  {
    const size_t yb = (size_t)b * N_ * 3;
    for (int t = threadIdx.x; t < N_; t += 256) {
      float x = Y[yb + 3 * t], y = Y[yb + 3 * t + 1], z = Y[yb + 3 * t + 2];
      sB0[t] = x; sB0[HSKEW + t] = z;
      sB1[t] = y; sB1[HSKEW + t] = 1.0f;
      sYs[t] = x * x + y * y + z * z;
      sYv[t] = ratioR[b * N_ + t];
    }
  }
  __syncthreads();

  const int lane = threadIdx.x & 31;
  const int w    = __builtin_amdgcn_readfirstlane(threadIdx.x >> 5);
  const int r    = lane & 15, h = lane >> 4;
  const int hoff = h * HSKEW;

  const size_t xb = ((size_t)b * N_ + (i0 + r)) * 3;
  const float px = X[xb], py = X[xb + 1], pz = X[xb + 2];
  const float ssx = px * px + py * py + pz * pz;
  v2f av; av.x = h ? -2.f * pz : -2.f * px;
          av.y = h ?  ssx      : -2.f * py;

  float rl[8];
#pragma unroll
  for (int k = 0; k < 8; ++k) rl[k] = ratioL[b * N_ + i0 + k + 8 * h];

  v8f accd = {};
  const v8f cz = {};
  float ccost = 0.f;
  const size_t mb = (size_t)b * N_ * N_;

  for (int s = 0; s < 16; ++s) {               // disjoint j slices: no races
    const int jn = w * 16 + s * 128 + r;
    v2f bv; bv.x = sB0[hoff + jn];
            bv.y = sB1[hoff + jn];
    v8f d = __builtin_amdgcn_wmma_f32_16x16x4_f32(
        false, av, false, bv, (short)0, cz, false, false);
    const float sq2   = sYs[jn];
    const float l2sq2 = l2 * sq2;
    const float vv    = sYv[jn];
#pragma unroll
    for (int k = 0; k < 8; ++k) {
      const float wgt = fast_exp2(__builtin_fmaf(l2, d[k], l2sq2));
      const float dd  = (wgt * vv) * rl[k];
      const size_t off = mb + (size_t)(i0 + k + 8 * h) * N_ + jn;
      const float nv = (first ? 0.f : match[off]) + dd;
      match[off] = nv;                          // 64B-coalesced per half-wave
      accd[k] += dd;
      if (last) {
        const float sq = d[k] + sq2;
        ccost += nv * sqrtf(fmaxf(sq, 0.f));
      }
    }
  }
#pragma unroll
  for (int k = 0; k < 8; ++k) accd[k] = half16_sum(accd[k]);
  if (r == 0) {
#pragma unroll
    for (int k = 0; k < 8; ++k) sRed[w][k + 8 * h] = accd[k];
  }
  if (last) {
    ccost += __shfl_xor(ccost, 16, 32);
    ccost = half16_sum(ccost);
    if (lane == 0) sCost[w] = ccost;
  }
  __syncthreads();

  if (threadIdx.x < 16) {
    const int row = threadIdx.x;
    float rs = 0.f;
#pragma unroll
    for (int q = 0; q < 8; ++q) rs += sRed[q][row];
    const int i = b * N_ + i0 + row;
    remL[i] = fmaxf(0.f, remL[i] - rs);
  }
  if (last && threadIdx.x == 0) {
    float cs = 0.f;
#pragma unroll
    for (int q = 0; q < 8; ++q) cs += sCost[q];
    atomicAdd(&cost[b], cs);
  }
}

// Race-free in-place transpose: each unordered tile pair {I,J} -> one block.
#define TT    32
#define TG    (N_ / TT)            // 64
#define PAIRS (TG * (TG + 1) / 2)  // 2080

__global__ __launch_bounds__(256)
void emd_transpose(float* __restrict__ mt) {
  const int b = blockIdx.x / PAIRS;
  int p = blockIdx.x % PAIRS;
  int I = 0;
  while (p >= TG - I) { p -= TG - I; ++I; }
  const int J = I + p;
  __shared__ float tA[TT][TT + 1];
  __shared__ float tB[TT][TT + 1];
  float* base = mt + (size_t)b * N_ * N_;
  const int c = threadIdx.x & 31, r0 = threadIdx.x >> 5;
  for (int r = r0; r < TT; r += 8) {
    tA[r][c] = base[(size_t)(I * TT + r) * N_ + J * TT + c];
    tB[r][c] = base[(size_t)(J * TT + r) * N_ + I * TT + c];
  }
  __syncthreads();
  for (int r = r0; r < TT; r += 8) {
    base[(size_t)(I * TT + r) * N_ + J * TT + c] = tB[c][r];
    base[(size_t)(J * TT + r) * N_ + I * TT + c] = tA[c][r];
  }
}

extern "C" void kernel_launch(void* const* d_in, const int* in_sizes, int n_in,
                              void* d_out, int out_size, void* d_ws, size_t ws_size,
                              hipStream_t stream) {
  (void)in_sizes; (void)n_in; (void)out_size; (void)ws_size;
  const float* xyz1 = (const float*)d_in[0];   // (8,2048,3) f32
  const float* xyz2 = (const float*)d_in[1];   // (8,2048,3) f32
  float* out   = (float*)d_out;
  float* cost  = out;                          // (8,)
  float* match = out + 8;                      // built (b,i,j), transposed at end
  float* ws     = (float*)d_ws;                // 4 * 16384 floats = 256 KB
  float* remL   = ws;
  float* remR   = ws + BN_;
  float* ratioL = ws + 2 * BN_;
  float* ratioR = ws + 3 * BN_;

  emd_init<<<(BN_ + 255) / 256, 256, 0, stream>>>(cost, remL, remR);

  const float levels[10] = {-16384.f, -4096.f, -1024.f, -256.f, -64.f,
                            -16.f,    -4.f,    -1.f,    -0.25f, 0.f};
  for (int t = 0; t < 10; ++t) {
    const float l2 = levels[t] * LOG2E_;       // exp(level*x) == exp2(l2*x)
    // pass 1: ratioL from remL, remR
    emd_ratio<<<PASSBLK, 256, 0, stream>>>(xyz1, xyz2, remR, remL,
                                           ratioL, nullptr, l2);
    // pass 2: ratioR from remR, ratioL; updates remR (delta col-sums are free)
    emd_ratio<<<PASSBLK, 256, 0, stream>>>(xyz2, xyz1, ratioL, remR,
                                           ratioR, remR, l2);
    // pass 3: match += delta, remL update, cost on last level
    emd_update<<<PASSBLK, 256, 0, stream>>>(xyz1, xyz2, ratioL, ratioR,
                                            remL, match, cost, l2,
                                            t == 0 ? 1 : 0, t == 9 ? 1 : 0);
  }
  emd_transpose<<<B_ * PAIRS, 256, 0, stream>>>(match);
}